// MultiBoxLoss_87162066305727
// MI455X (gfx1250) — compile-verified
//
#include <hip/hip_runtime.h>
#include <stdint.h>

#define BB 32
#define PP 16384
#define NOBJ 50
#define CC 21
#define THRESH 0.5f

#define TILE 256
#define NBLK (BB * (PP / TILE)) // 2048

typedef float v2f __attribute__((ext_vector_type(2)));
typedef float v8f __attribute__((ext_vector_type(8)));

// ---------------- CDNA5 helpers ----------------

__device__ __forceinline__ uint32_t lds_off(const void* p) {
  // generic shared pointer: low 32 bits are the LDS offset (HW adds LDS_BASE)
  return (uint32_t)(uintptr_t)p;
}

__device__ __forceinline__ void async_b128(uint32_t dst, const void* g) {
  asm volatile("global_load_async_to_lds_b128 %0, %1, off"
               :: "v"(dst), "v"(g) : "memory");
}
__device__ __forceinline__ void async_b32(uint32_t dst, const void* g) {
  asm volatile("global_load_async_to_lds_b32 %0, %1, off"
               :: "v"(dst), "v"(g) : "memory");
}
__device__ __forceinline__ void wait_async() {
  asm volatile("s_wait_asynccnt 0" ::: "memory");
}

// Full 32-lane sum using V_WMMA_F32_16X16X4_F32.
// A[m,k]: lanes0-15 hold (K0,K1) for row m=lane; lanes16-31 hold (K2,K3) for m=lane-16.
// With a=(s,0), B=ones(4x16): D[m,n] = s[m] + s[m+16].
// D layout: lane n<16 holds D[0..7,n]; lane n>=16 holds D[8..15,n].
// So sum of d[0..7] on lane0 = sum_{m=0..7}(s_m+s_{m+16}); lane16 covers m=8..15.
__device__ __forceinline__ float wave_sum_wmma(float s) {
  v2f a; a[0] = s;    a[1] = 0.0f;
  v2f b; b[0] = 1.0f; b[1] = 1.0f;
  v8f c = {0.f, 0.f, 0.f, 0.f, 0.f, 0.f, 0.f, 0.f};
  v8f d = __builtin_amdgcn_wmma_f32_16x16x4_f32(false, a, false, b,
                                                (short)0, c, false, false);
  float t = d[0] + d[1] + d[2] + d[3] + d[4] + d[5] + d[6] + d[7];
  return __shfl(t, 0, 32) + __shfl(t, 16, 32);
}

// ---------------- kernels ----------------

__global__ void k_init(unsigned long long* best64, int* nPos) {
  int i = blockIdx.x * blockDim.x + threadIdx.x;
  if (i < BB * NOBJ) best64[i] = 0ull;
  if (i < BB) nPos[i] = 0;
}

__global__ __launch_bounds__(TILE)
void k_assign(const float* __restrict__ priors, const float* __restrict__ boxes,
              float* __restrict__ ovmax, int* __restrict__ objfor,
              unsigned long long* __restrict__ best64) {
  __shared__ float sBox[NOBJ * 4];
  __shared__ float sPri[TILE * 4];
  __shared__ unsigned long long sBest[NOBJ];

  const int tid = threadIdx.x;
  const int b = blockIdx.x / (PP / TILE);
  const int chunk = blockIdx.x % (PP / TILE);
  const int p = chunk * TILE + tid;

  if (tid < NOBJ) {
    sBest[tid] = 0ull;
    async_b128(lds_off(&sBox[tid * 4]), boxes + (size_t)b * NOBJ * 4 + tid * 4);
  }
  async_b128(lds_off(&sPri[tid * 4]), priors + (size_t)p * 4);
  wait_async();
  __syncthreads();

  const float cx = sPri[tid * 4 + 0], cy = sPri[tid * 4 + 1];
  const float w  = sPri[tid * 4 + 2], h  = sPri[tid * 4 + 3];
  const float px1 = cx - w * 0.5f, py1 = cy - h * 0.5f;
  const float px2 = cx + w * 0.5f, py2 = cy + h * 0.5f;
  const float areaP = w * h;

  float bestIou = -1.0f;
  int bestN = 0;
  for (int n = 0; n < NOBJ; ++n) {
    const float x1 = sBox[n * 4 + 0], y1 = sBox[n * 4 + 1];
    const float x2 = sBox[n * 4 + 2], y2 = sBox[n * 4 + 3];
    const float lx = fmaxf(x1, px1), ly = fmaxf(y1, py1);
    const float rx = fminf(x2, px2), ry = fminf(y2, py2);
    const float iw = fmaxf(rx - lx, 0.f), ih = fmaxf(ry - ly, 0.f);
    const float inter = iw * ih;
    const float areaA = (x2 - x1) * (y2 - y1);
    const float iou = inter / (areaA + areaP - inter);
    if (iou > bestIou) { bestIou = iou; bestN = n; } // first-max => lowest n
    // per-object argmax over priors; ties pick smallest p (matches jnp.argmax)
    const unsigned long long key =
        ((unsigned long long)__float_as_uint(iou) << 32) | (unsigned int)(~p);
    atomicMax(&sBest[n], key);
  }
  ovmax [(size_t)b * PP + p] = bestIou;
  objfor[(size_t)b * PP + p] = bestN;
  __syncthreads();
  if (tid < NOBJ) atomicMax(&best64[b * NOBJ + tid], sBest[tid]);
}

__global__ void k_fixup(const unsigned long long* __restrict__ best64,
                        float* __restrict__ ovmax, int* __restrict__ objfor) {
  const int b = threadIdx.x;
  if (b >= BB) return;
  for (int n = 0; n < NOBJ; ++n) { // serial: later object wins on duplicates
    const unsigned long long k = best64[b * NOBJ + n];
    const unsigned int p = ~(unsigned int)(k & 0xffffffffu);
    ovmax [(size_t)b * PP + p] = 1.0f;
    objfor[(size_t)b * PP + p] = n;
  }
}

__global__ __launch_bounds__(TILE)
void k_main(const float* __restrict__ ploc, const float* __restrict__ pscore,
            const float* __restrict__ boxes, const int* __restrict__ labels,
            const float* __restrict__ priors,
            const float* __restrict__ ovmax, const int* __restrict__ objfor,
            float* __restrict__ neg, int* __restrict__ nPos,
            float* __restrict__ partL1, float* __restrict__ partCP) {
  __shared__ float sBox[NOBJ * 4];
  __shared__ float sPri[TILE * 4];
  __shared__ float sSc[TILE * CC];
  __shared__ float sWL1[TILE / 32], sWCP[TILE / 32];
  __shared__ int sNp;

  const int tid = threadIdx.x;
  const int b = blockIdx.x / (PP / TILE);
  const int chunk = blockIdx.x % (PP / TILE);
  const int p0 = chunk * TILE;
  const int p = p0 + tid;

  if (tid == 0) sNp = 0;
  if (tid < NOBJ)
    async_b128(lds_off(&sBox[tid * 4]), boxes + (size_t)b * NOBJ * 4 + tid * 4);
  async_b128(lds_off(&sPri[tid * 4]), priors + (size_t)p * 4);
  const float* scBase = pscore + ((size_t)b * PP + p0) * CC;
  for (int i = tid; i < TILE * CC; i += TILE)
    async_b32(lds_off(&sSc[i]), scBase + i);
  wait_async();
  __syncthreads();

  const int obj = objfor[(size_t)b * PP + p];
  const float ov = ovmax[(size_t)b * PP + p];
  int lab = labels[b * NOBJ + obj];
  if (ov < THRESH) lab = 0;
  const bool pos = (lab != 0);

  // cross entropy via log-sum-exp over 21 classes (staged in LDS)
  const float* sc = &sSc[tid * CC];
  float m = sc[0];
  for (int c2 = 1; c2 < CC; ++c2) m = fmaxf(m, sc[c2]);
  float sum = 0.f;
  for (int c2 = 0; c2 < CC; ++c2) sum += expf(sc[c2] - m);
  const float ce = m + logf(sum) - sc[lab];

  neg[(size_t)b * PP + p] = pos ? 0.f : ce;
  const float cp = pos ? ce : 0.f;

  // SSD box encoding + masked L1
  float l1 = 0.f;
  {
    const float x1 = sBox[obj * 4 + 0], y1 = sBox[obj * 4 + 1];
    const float x2 = sBox[obj * 4 + 2], y2 = sBox[obj * 4 + 3];
    const float bcx = (x1 + x2) * 0.5f, bcy = (y1 + y2) * 0.5f;
    const float bw = x2 - x1, bh = y2 - y1;
    const float cx = sPri[tid * 4 + 0], cy = sPri[tid * 4 + 1];
    const float w  = sPri[tid * 4 + 2], h  = sPri[tid * 4 + 3];
    const float g0 = (bcx - cx) * 10.f / w;
    const float g1 = (bcy - cy) * 10.f / h;
    const float g2 = logf(bw / w) * 5.f;
    const float g3 = logf(bh / h) * 5.f;
    const float* pl = ploc + ((size_t)b * PP + p) * 4;
    l1 = fabsf(pl[0] - g0) + fabsf(pl[1] - g1) +
         fabsf(pl[2] - g2) + fabsf(pl[3] - g3);
    if (!pos) l1 = 0.f;
  }

  // deterministic reductions: WMMA wave sums -> fixed block slots
  const float wl1 = wave_sum_wmma(l1);
  const float wcp = wave_sum_wmma(cp);
  const int wnp = __popcll(__ballot(pos));
  const int lane = tid & 31, wv = tid >> 5;
  if (lane == 0) { sWL1[wv] = wl1; sWCP[wv] = wcp; atomicAdd(&sNp, wnp); }
  __syncthreads();
  if (tid == 0) {
    float a = 0.f, c = 0.f;
    for (int i = 0; i < TILE / 32; ++i) { a += sWL1[i]; c += sWCP[i]; }
    partL1[blockIdx.x] = a;
    partCP[blockIdx.x] = c;
    atomicAdd(&nPos[b], sNp); // int: order-independent
  }
}

__global__ __launch_bounds__(1024)
void k_hard(const float* __restrict__ neg, const int* __restrict__ nPos,
            float* __restrict__ hardSum) {
  __shared__ float s[PP]; // 64 KB: one image's negatives, sorted in-LDS
  const int b = blockIdx.x;
  const int tid = threadIdx.x;
  for (int i = tid; i < PP; i += 1024) s[i] = neg[(size_t)b * PP + i];
  __syncthreads();

  // bitonic sort, descending
  for (int k = 2; k <= PP; k <<= 1) {
    for (int j = k >> 1; j > 0; j >>= 1) {
      for (int t = tid; t < PP / 2; t += 1024) {
        const int i = 2 * t - (t & (j - 1));
        const int ixj = i + j;
        const bool up = ((i & k) == 0);
        const float a = s[i], c = s[ixj];
        if (up ? (a < c) : (a > c)) { s[i] = c; s[ixj] = a; }
      }
      __syncthreads();
    }
  }

  int K = 3 * nPos[b];
  if (K > PP) K = PP;
  float t = 0.f;
  for (int i = tid; i < K; i += 1024) t += s[i];
  __syncthreads();
  s[tid] = t;
  __syncthreads();
  for (int st = 512; st > 0; st >>= 1) {
    if (tid < st) s[tid] += s[tid + st];
    __syncthreads();
  }
  if (tid == 0) hardSum[b] = s[0];
}

__global__ void k_final(const float* __restrict__ partL1,
                        const float* __restrict__ partCP,
                        const int* __restrict__ nPos,
                        const float* __restrict__ hardSum,
                        float* __restrict__ out) {
  if (threadIdx.x != 0 || blockIdx.x != 0) return;
  float l1 = 0.f, cp = 0.f, hn = 0.f;
  long np = 0;
  for (int i = 0; i < NBLK; ++i) { l1 += partL1[i]; cp += partCP[i]; }
  for (int b = 0; b < BB; ++b) { np += nPos[b]; hn += hardSum[b]; }
  const float npf = (float)np;
  out[0] = (hn + cp) / npf + l1 / (npf * 4.0f);
}

// ---------------- launch ----------------

extern "C" void kernel_launch(void* const* d_in, const int* in_sizes, int n_in,
                              void* d_out, int out_size, void* d_ws, size_t ws_size,
                              hipStream_t stream) {
  (void)in_sizes; (void)n_in; (void)out_size; (void)ws_size;
  const float* ploc   = (const float*)d_in[0]; // [B,P,4]
  const float* pscore = (const float*)d_in[1]; // [B,P,C]
  const float* boxes  = (const float*)d_in[2]; // [B,NOBJ,4]
  const int*   labels = (const int*)  d_in[3]; // [B,NOBJ]
  const float* priors = (const float*)d_in[4]; // [P,4]

  char* ws = (char*)d_ws;
  size_t off = 0;
  auto carve = [&](size_t bytes) -> void* {
    void* q = ws + off;
    off += (bytes + 255) & ~(size_t)255;
    return q;
  };
  unsigned long long* best64 = (unsigned long long*)carve(BB * NOBJ * 8);
  float* ovmax   = (float*)carve((size_t)BB * PP * 4);
  int*   objfor  = (int*)  carve((size_t)BB * PP * 4);
  float* neg     = (float*)carve((size_t)BB * PP * 4);
  int*   nPos    = (int*)  carve(BB * 4);
  float* partL1  = (float*)carve(NBLK * 4);
  float* partCP  = (float*)carve(NBLK * 4);
  float* hardSum = (float*)carve(BB * 4);

  k_init  <<<(BB * NOBJ + 255) / 256, 256, 0, stream>>>(best64, nPos);
  k_assign<<<NBLK, TILE, 0, stream>>>(priors, boxes, ovmax, objfor, best64);
  k_fixup <<<1, BB, 0, stream>>>(best64, ovmax, objfor);
  k_main  <<<NBLK, TILE, 0, stream>>>(ploc, pscore, boxes, labels, priors,
                                      ovmax, objfor, neg, nPos, partL1, partCP);
  k_hard  <<<BB, 1024, 0, stream>>>(neg, nPos, hardSum);
  k_final <<<1, 1, 0, stream>>>(partL1, partCP, nPos, hardSum, (float*)d_out);
}